// CustomLoss_19971597926550
// MI455X (gfx1250) — compile-verified
//
#include <hip/hip_runtime.h>
#include <stdint.h>

// CustomLoss: softmax top-2 scoring reduction over [1048576, 32] f32.
// Memory-bound: 132 MB -> scalar; roofline ~5.7us @ 23.3 TB/s.
// CDNA5 path: async global->LDS staging (ASYNCcnt) + padded LDS rows,
// thread-per-row compute with native v_exp_f32.

#define TPB   256
#define NBLK  1024
#define ROWC  32                 // classes per row (128B = 1 cacheline)
#define PADW  36                 // padded LDS row stride in words (144B, bank-conflict-free b128)
#define L2E   1.44269504088896340736f

#if defined(__AMDGCN__) && __has_builtin(__builtin_amdgcn_global_load_async_to_lds_b128) && __has_builtin(__builtin_amdgcn_s_wait_asynccnt)
#define USE_ASYNC_LDS 1
#else
#define USE_ASYNC_LDS 0
#endif

#define AS1 __attribute__((address_space(1)))
#define AS3 __attribute__((address_space(3)))

typedef int v4i_ __attribute__((__vector_size__(4 * sizeof(int))));

#if USE_ASYNC_LDS
// global -> LDS async 16B copy; builtin signature (probe-confirmed via
// diagnostic): (v4i AS1*, v4i AS3*, imm offset, imm cpol)
__device__ __forceinline__ void async_copy_b128(const float* g, float* l) {
  __builtin_amdgcn_global_load_async_to_lds_b128(
      (AS1 v4i_*)(uintptr_t)g,
      (AS3 v4i_*)(uint32_t)(uintptr_t)l,
      0, 0);
}
#endif

__device__ __forceinline__ float row_score(const float x[32], float xt) {
  // top-1 / top-2 of logits (softmax is monotone)
  float m1 = x[0];
  float m2 = -3.402823466e+38f;
#pragma unroll
  for (int i = 1; i < 32; ++i) {
    float v  = x[i];
    float hi = fmaxf(m1, v);
    float lo = fminf(m1, v);
    m1 = hi;
    m2 = fmaxf(m2, lo);
  }
  // sum of exp(xi - m1) via native exp2
  float nk = -m1 * L2E;
  float s  = 0.0f;
#pragma unroll
  for (int i = 0; i < 32; ++i)
    s += __builtin_amdgcn_exp2f(fmaf(x[i], L2E, nk));
  float rs   = 1.0f / s;
  float e2   = __builtin_amdgcn_exp2f(fmaf(m2, L2E, nk));
  float diff = (1.0f - e2) * rs;          // p_top1 - p_top2 >= 0
  bool hit1  = (xt == m1);
  bool hit2  = (!hit1) && (xt == m2);
  float amb  = hit1 ? 0.8f : (hit2 ? 0.6f : 0.0f);
  float conf = hit1 ? 1.0f : 0.0f;
  return (diff < 0.5f) ? amb : conf;
}

__global__ __launch_bounds__(TPB) void loss_partial_kernel(
    const float* __restrict__ inp, const int* __restrict__ tgt,
    float* __restrict__ partial, int B) {
  const int tid = threadIdx.x;
  float acc = 0.0f;

#if USE_ASYNC_LDS
  __shared__ __align__(16) float lds[2][TPB * PADW];   // 2 x 36KB double buffer
  const int ntiles  = (B + TPB - 1) / TPB;
  const size_t maxw = (size_t)B * ROWC - 4;            // last valid b128 word index

  auto stage = [&](int tile, int b) {
    // 8 async b128 per thread: global side perfectly coalesced (c*16B),
    // LDS side row-padded to 144B.
#pragma unroll
    for (int j = 0; j < 8; ++j) {
      int c = j * TPB + tid;                            // 16B chunk id in tile
      size_t gw = (size_t)tile * (TPB * ROWC) + (size_t)c * 4;
      if (gw > maxw) gw = 0;                            // clamp (tail tiles): garbage rows never read
      int lw = (c >> 3) * PADW + (c & 7) * 4;
      async_copy_b128(inp + gw, &lds[b][lw]);
    }
  };

  int tile = blockIdx.x;
  int b = 0;
  if (tile < ntiles) stage(tile, 0);
  for (; tile < ntiles; tile += (int)gridDim.x, b ^= 1) {
    int next = tile + (int)gridDim.x;
    if (next < ntiles) {
      stage(next, b ^ 1);                               // prefetch next tile into other buffer
      __builtin_amdgcn_s_wait_asynccnt(8);              // current tile's 8 ops complete (in-order)
    } else {
      __builtin_amdgcn_s_wait_asynccnt(0);
    }
    __syncthreads();                                    // all waves' async ops for this tile done

    int row = tile * TPB + tid;
    if (row < B) {
      float x[32];
      const float4* v4 = (const float4*)&lds[b][tid * PADW];  // conflict-free ds_load_b128
#pragma unroll
      for (int j = 0; j < 8; ++j) {
        float4 q = v4[j];
        x[4 * j + 0] = q.x; x[4 * j + 1] = q.y;
        x[4 * j + 2] = q.z; x[4 * j + 3] = q.w;
      }
      int tg   = tgt[row];
      float xt = lds[b][tid * PADW + tg];
      acc += row_score(x, xt);
    }
    __syncthreads();                                    // buffer b free for restage next iter
  }
#else
  // Fallback: direct b128 loads (each row = exactly one 128B line) + prefetch
  for (int row = blockIdx.x * TPB + tid; row < B; row += (int)gridDim.x * TPB) {
    const float4* v4 = (const float4*)(inp + (size_t)row * ROWC);
    __builtin_prefetch(inp + ((size_t)row + (size_t)gridDim.x * TPB) * ROWC, 0, 0);
    float x[32];
#pragma unroll
    for (int j = 0; j < 8; ++j) {
      float4 q = v4[j];
      x[4 * j + 0] = q.x; x[4 * j + 1] = q.y;
      x[4 * j + 2] = q.z; x[4 * j + 3] = q.w;
    }
    int tg   = tgt[row];
    float xt = inp[(size_t)row * ROWC + tg];
    acc += row_score(x, xt);
  }
#endif

  // deterministic block reduction
  __shared__ float red[TPB];
  red[tid] = acc;
  __syncthreads();
#pragma unroll
  for (int s = TPB >> 1; s > 0; s >>= 1) {
    if (tid < s) red[tid] += red[tid + s];
    __syncthreads();
  }
  if (tid == 0) partial[blockIdx.x] = red[0];
}

__global__ __launch_bounds__(TPB) void loss_finalize_kernel(
    const float* __restrict__ partial, int n, float* __restrict__ out) {
  __shared__ float red[TPB];
  float a = 0.0f;
  for (int i = threadIdx.x; i < n; i += TPB) a += partial[i];
  red[threadIdx.x] = a;
  __syncthreads();
#pragma unroll
  for (int s = TPB >> 1; s > 0; s >>= 1) {
    if (threadIdx.x < s) red[threadIdx.x] += red[threadIdx.x + s];
    __syncthreads();
  }
  if (threadIdx.x == 0) out[0] = -red[0];
}

extern "C" void kernel_launch(void* const* d_in, const int* in_sizes, int n_in,
                              void* d_out, int out_size, void* d_ws, size_t ws_size,
                              hipStream_t stream) {
  (void)n_in; (void)out_size;
  const float* inp = (const float*)d_in[0];   // [B, 32] f32
  const int*   tgt = (const int*)d_in[1];     // [B] i32
  const int B = in_sizes[1];

  int nblk = NBLK;
  size_t maxp = ws_size / sizeof(float);
  if ((size_t)nblk > maxp) nblk = (int)(maxp ? maxp : 1);

  float* partial = (float*)d_ws;
  loss_partial_kernel<<<nblk, TPB, 0, stream>>>(inp, tgt, partial, B);
  loss_finalize_kernel<<<1, TPB, 0, stream>>>(partial, nblk, (float*)d_out);
}